// Successive_over_relaxation_45999099740609
// MI455X (gfx1250) — compile-verified
//
#include <hip/hip_runtime.h>
#include <math.h>

// ---------------------------------------------------------------------------
// Red/black SOR Laplace solver (130^3 padded grid), CDNA5 / gfx1250.
//  - halo tile (34x6x6 f32) DMA'd into LDS by the Tensor Data Mover
//    (tensor_load_to_lds, TENSORcnt) -- one instruction per block
//  - interior mask bytes staged concurrently via GLOBAL_LOAD_ASYNC_TO_LDS_B8
//    (ASYNCcnt); split wait counters let both transfers overlap
//  - fixed 501-iteration launch schedule (graph-capture safe), device-side
//    convergence gating, deterministic tree reductions (no float atomics)
// ---------------------------------------------------------------------------

#define PN    130                 // padded side
#define PNN   (PN * PN)
#define PVOL  (PN * PN * PN)      // 2,197,000
#define IVOL  (128 * 128 * 128)   // 2,097,152
#define WOPT  ((float)(2.0 / (1.0 + 3.14159265358979323846 / 130.0)))
#define THRESH 0.05f
#define MAX_IT 500
#define NBLK   4096               // 4 x 32 x 32 tile blocks

#define TILE_X 34                 // halo box: 34 x 6 x 6 floats (packed rows)
#define TILE_N (6 * 6 * TILE_X)

struct SorCtl {
    int   it;
    float dv;
    int   converged;
    int   _pad;
};

typedef unsigned int       u32;
typedef unsigned long long u64;
typedef u32 v4u __attribute__((ext_vector_type(4)));
typedef int v4i __attribute__((ext_vector_type(4)));
typedef int v8i __attribute__((ext_vector_type(8)));

#if __has_builtin(__builtin_amdgcn_tensor_load_to_lds)
#define SOR_HAS_TDM 1
#else
#define SOR_HAS_TDM 0
#endif

// --- CDNA5 async global->LDS copies (per-lane) ------------------------------
// Generic LDS pointers map to LDS via addr[31:0] (ISA 10.2), so truncation
// yields the wave-relative LDS byte offset the instructions expect.
__device__ __forceinline__ void async_g2l_b32(unsigned lds_byte, const float* g) {
    unsigned long long ga = (unsigned long long)(size_t)g;
    asm volatile("global_load_async_to_lds_b32 %0, %1, off"
                 :: "v"(lds_byte), "v"(ga) : "memory");
}
__device__ __forceinline__ void async_g2l_b8(unsigned lds_byte, const unsigned char* g) {
    unsigned long long ga = (unsigned long long)(size_t)g;
    asm volatile("global_load_async_to_lds_b8 %0, %1, off"
                 :: "v"(lds_byte), "v"(ga) : "memory");
}
__device__ __forceinline__ void wait_async_zero() {
    asm volatile("s_wait_asynccnt 0" ::: "memory");
}

#if SOR_HAS_TDM
// --- Tensor Data Mover: DMA a 34x6x6 f32 tile from a 130^3 tensor to LDS ---
// Descriptor built per CDNA5 ISA section 8 (D# groups 0..3), 3D tile:
//   tile_dim  = {34, 6, 6}, data_size = 4B (code 2)
//   strides   = {130, 16900} (data_size units), tensor dims 130^3
__device__ __forceinline__ void tdm_load_tile(unsigned lds_byte, const float* gptr)
{
    u64 ga = (u64)(size_t)gptr;

    v4u g0;
    g0.x = 1u;                                         // count=1, user D#
    g0.y = lds_byte;                                   // lds_addr
    g0.z = (u32)(ga & 0xFFFFFFFFu);                    // global_addr[31:0]
    g0.w = (u32)((ga >> 32) & 0x01FFFFFFu)             // global_addr[56:32]
         | (2u << 30);                                 // type = 2 ("image")

    const u64 td0 = PN,  td1 = PN;                     // tensor_dim0/1
    const u64 s0  = PN,  s1  = PNN;                    // dim0/dim1 strides
    const u64 t0  = TILE_X, t1 = 6, t2 = 6;            // tile dims

    // group1, 256 bits, assembled as 4 x u64 per the bit layout
    u64 q0 = ((u64)2u << 16)                           // data_size = 4 bytes
           | ((td0 & 0xFFFFull) << 48);                // tensor_dim0[15:0]  @48
    u64 q1 = (td0 >> 16)                               // tensor_dim0[31:16] @64
           | (td1 << 16)                               // tensor_dim1       @80
           | (t0  << 48);                              // tile_dim0         @112
    u64 q2 = t1                                        // tile_dim1         @128
           | (t2 << 16)                                // tile_dim2         @144
           | ((s0 & 0xFFFFFFFFull) << 32);             // stride0[31:0]     @160
    u64 q3 = (s0 >> 32)                                // stride0[47:32]    @192
           | (s1 << 16);                               // stride1           @208

    v8i g1;
    g1[0] = (int)(u32)q0; g1[1] = (int)(u32)(q0 >> 32);
    g1[2] = (int)(u32)q1; g1[3] = (int)(u32)(q1 >> 32);
    g1[4] = (int)(u32)q2; g1[5] = (int)(u32)(q2 >> 32);
    g1[6] = (int)(u32)q3; g1[7] = (int)(u32)(q3 >> 32);

    v4i g2 = { PN, 0, 0, 0 };   // tensor_dim2 = 130; dim3/stride2/tile_dim3 = 0
    v4i g3 = { 0, 0, 0, 0 };

#if defined(__clang_major__) && (__clang_major__ >= 23)
    v8i g4 = { 0, 0, 0, 0, 0, 0, 0, 0 };
    __builtin_amdgcn_tensor_load_to_lds(g0, g1, g2, g3, g4, 0);
#else
    __builtin_amdgcn_tensor_load_to_lds(g0, g1, g2, g3, 0);
#endif
}
#endif // SOR_HAS_TDM

// --- init: build v (sink->1 else 0) and mask (0 inactive, 1 black, 2 red) ---
__global__ void sor_init(const float* __restrict__ img,
                         float* __restrict__ v,
                         unsigned char* __restrict__ mask,
                         SorCtl* __restrict__ ctl,
                         const int* __restrict__ sink_label)
{
    int idx = blockIdx.x * blockDim.x + threadIdx.x;
    if (idx == 0) {
        ctl->it = 0;
        ctl->dv = 0.0f;
        ctl->converged = 0;
    }
    if (idx >= PVOL) return;

    int k = idx % PN;
    int j = (idx / PN) % PN;
    int i = idx / PNN;

    float val = 0.0f;   // zero padding
    if (i >= 1 && i <= 128 && j >= 1 && j <= 128 && k >= 1 && k <= 128)
        val = img[((i - 1) * 128 + (j - 1)) * 128 + (k - 1)];

    float snk = (float)(*sink_label);
    v[idx] = (val == snk) ? 1.0f : 0.0f;
    // parity on PADDED coords, restricted to grey matter (label==1)
    mask[idx] = (val == 1.0f) ? (unsigned char)(1 + ((i + j + k) & 1)) : (unsigned char)0;
}

// --- one half-sweep over a 32x4x4 tile --------------------------------------
__global__ void sor_sweep(float* __restrict__ v,
                          const unsigned char* __restrict__ mask,
                          const int* __restrict__ converged,
                          float* __restrict__ partial,
                          int parity_target /* 1 = black, 2 = red */)
{
    if (*converged) return;  // uniform branch: safe before barriers

    __shared__ float         tile[TILE_N];   // packed rows of 34 floats
    __shared__ unsigned char mtile[512];
    __shared__ float         redbuf[256];

    const int tid = threadIdx.x;
    // interior tile origin in padded coords
    const int ox = 1 + 32 * blockIdx.x;
    const int oy = 1 + 4  * blockIdx.y;
    const int oz = 1 + 4  * blockIdx.z;
    const int gstart = ((oz - 1) * PN + (oy - 1)) * PN + (ox - 1);

#if SOR_HAS_TDM
    if (tid < 32) {  // one wave drives the TDM engine for the whole block
        tdm_load_tile((unsigned)(size_t)&tile[0], v + gstart);
    }
#else
    for (int e = tid; e < TILE_N; e += 256) {
        int lz = e / (6 * TILE_X);
        int r  = e % (6 * TILE_X);
        int ly = r / TILE_X;
        int lx = r % TILE_X;
        int g  = ((oz - 1 + lz) * PN + (oy - 1 + ly)) * PN + (ox - 1 + lx);
        async_g2l_b32((unsigned)(size_t)&tile[e], v + g);
    }
#endif

    // stage interior mask bytes concurrently with the tile DMA (ASYNCcnt)
    for (int c = tid; c < 512; c += 256) {
        int lx = c & 31, ly = (c >> 5) & 3, lz = c >> 7;
        int g = ((oz + lz) * PN + (oy + ly)) * PN + (ox + lx);
        async_g2l_b8((unsigned)(size_t)&mtile[c], mask + g);
    }

#if SOR_HAS_TDM
    if (tid < 32) __builtin_amdgcn_s_wait_tensorcnt(0);
#endif
    wait_async_zero();
    __syncthreads();

    // ---- update: 512 cells / block, 2 per thread ----
    float local_abs = 0.0f;
#pragma unroll
    for (int c = tid; c < 512; c += 256) {
        int lx = c & 31;
        int ly = (c >> 5) & 3;
        int lz = c >> 7;
        int px = ox + lx, py = oy + ly, pz = oz + lz;
        int g  = (pz * PN + py) * PN + px;

        unsigned char m = mtile[c];
        int ci = ((lz + 1) * 6 + (ly + 1)) * TILE_X + (lx + 1);
        float center = tile[ci];
        float s = tile[ci - 1] + tile[ci + 1]
                + tile[ci - TILE_X] + tile[ci + TILE_X]
                + tile[ci - 6 * TILE_X] + tile[ci + 6 * TILE_X];

        bool active = (m == (unsigned char)parity_target);
        float adj = active ? (WOPT * (s - 6.0f * center) * (1.0f / 6.0f)) : 0.0f;
        if (active) v[g] = center + adj;   // in-place: neighbors are opposite color
        local_abs += fabsf(adj);
    }

    // ---- deterministic block reduction of sum|adj| ----
    redbuf[tid] = local_abs;
    __syncthreads();
    for (int sgap = 128; sgap > 0; sgap >>= 1) {
        if (tid < sgap) redbuf[tid] += redbuf[tid + sgap];
        __syncthreads();
    }
    if (tid == 0) {
        int b = (blockIdx.z * gridDim.y + blockIdx.y) * gridDim.x + blockIdx.x;
        partial[b] = redbuf[0];
    }
}

// --- end-of-iteration: reduce partials, apply while-loop condition ----------
__global__ void sor_finalize(SorCtl* __restrict__ ctl,
                             const float* __restrict__ pB,
                             const float* __restrict__ pR,
                             int n_part)
{
    if (ctl->converged) return;
    __shared__ float redbuf[256];
    const int tid = threadIdx.x;

    float s = 0.0f;
    for (int i = tid; i < n_part; i += 256) s += pB[i] + pR[i];
    redbuf[tid] = s;
    __syncthreads();
    for (int sgap = 128; sgap > 0; sgap >>= 1) {
        if (tid < sgap) redbuf[tid] += redbuf[tid + sgap];
        __syncthreads();
    }
    if (tid == 0) {
        float dv = redbuf[0];
        int itn = ctl->it + 1;
        ctl->it = itn;
        ctl->dv = dv;
        // reference cond (checked before next body): dv >= thresh && it <= 500
        if (!(dv >= THRESH && itn <= MAX_IT)) ctl->converged = 1;
    }
}

// --- gather interior + scalars into d_out -----------------------------------
__global__ void sor_output(const float* __restrict__ v,
                           const SorCtl* __restrict__ ctl,
                           float* __restrict__ out)
{
    int n = blockIdx.x * blockDim.x + threadIdx.x;
    if (n < IVOL) {
        int k = n & 127;
        int j = (n >> 7) & 127;
        int i = n >> 14;
        out[n] = v[((i + 1) * PN + (j + 1)) * PN + (k + 1)];
    }
    if (n == 0) {
        out[IVOL]     = (float)ctl->it;
        out[IVOL + 1] = ctl->dv;
    }
}

// ---------------------------------------------------------------------------
extern "C" void kernel_launch(void* const* d_in, const int* in_sizes, int n_in,
                              void* d_out, int out_size, void* d_ws, size_t ws_size,
                              hipStream_t stream)
{
    const float* img  = (const float*)d_in[0];
    // d_in[1] = source_label (unused: sources stay 0 and are never gm cells)
    const int*   snk  = (const int*)d_in[2];
    float*       out  = (float*)d_out;

    char* ws = (char*)d_ws;
    // workspace layout (~11 MB):
    SorCtl*        ctl  = (SorCtl*)ws;                                 // [0, 64)
    float*         pB   = (float*)(ws + 64);                           // 4096 f32
    float*         pR   = (float*)(ws + 64 + NBLK * 4);                // 4096 f32
    float*         v    = (float*)(ws + 33024);                        // 130^3 f32
    unsigned char* mask = (unsigned char*)(ws + 33024 + (size_t)PVOL * 4); // 130^3 u8

    sor_init<<<(PVOL + 255) / 256, 256, 0, stream>>>(img, v, mask, ctl, snk);

    dim3 grid(4, 32, 32);   // 32x4x4 tiles over the 128^3 interior
    for (int i = 0; i <= MAX_IT; ++i) {  // max 501 loop bodies
        sor_sweep<<<grid, 256, 0, stream>>>(v, mask, &ctl->converged, pB, 1); // black
        sor_sweep<<<grid, 256, 0, stream>>>(v, mask, &ctl->converged, pR, 2); // red
        sor_finalize<<<1, 256, 0, stream>>>(ctl, pB, pR, NBLK);
    }

    sor_output<<<(IVOL + 255) / 256, 256, 0, stream>>>(v, ctl, out);
}